// GraphCN_36240934043948
// MI455X (gfx1250) — compile-verified
//
#include <hip/hip_runtime.h>
#include <math.h>

typedef __attribute__((ext_vector_type(2))) float v2f;
typedef __attribute__((ext_vector_type(8))) float v8f;

#define N_NODES 100000
#define N_EDGES 1600000
#define FEAT    128
#define NGRAPH  64

// ---------------------------------------------------------------------------
// Utility fills
// ---------------------------------------------------------------------------
__global__ void fill_f32(float* __restrict__ p, float v, int n) {
    int i = blockIdx.x * blockDim.x + threadIdx.x;
    if (i < n) p[i] = v;
}

// deg[col[e]] += 1  (dinv buffer pre-filled with 1.0 for the self loop)
__global__ void deg_count(const int* __restrict__ col, float* __restrict__ deg, int e) {
    int i = blockIdx.x * blockDim.x + threadIdx.x;
    if (i < e) atomicAdd(&deg[col[i]], 1.0f);
}

__global__ void deg_to_dinv(float* __restrict__ deg, int n) {
    int i = blockIdx.x * blockDim.x + threadIdx.x;
    if (i < n) deg[i] = rsqrtf(deg[i]);   // deg >= 1 always (self loop)
}

// ---------------------------------------------------------------------------
// Dense projection: out[N x 128] = in[N x 128] @ W[128 x 128] + b
// WMMA f32 16x16x4, wave32. Block = 128 threads = 4 waves; block owns a
// 16-row stripe; wave w owns columns [32w, 32w+32) as two 16-col tiles.
// ---------------------------------------------------------------------------
__global__ __launch_bounds__(128)
void gcn_gemm_wmma(const float* __restrict__ in, const float* __restrict__ W,
                   const float* __restrict__ bias, float* __restrict__ out) {
    // stride 132 floats: bank = (4*lane + k) mod 64 -> conflict-free A reads
    __shared__ float atile[16][132];

    const int tid  = threadIdx.x;
    const int wave = tid >> 5;
    const int lane = tid & 31;
    const long rowbase = (long)blockIdx.x * 16;

    // Cooperative load of the 16x128 A tile (each thread: 16 contiguous floats)
    {
        const int r  = tid >> 3;
        const int c0 = (tid & 7) * 16;
        const float* src = in + (rowbase + r) * FEAT + c0;
        #pragma unroll
        for (int i = 0; i < 16; ++i) atile[r][c0 + i] = src[i];
    }
    __syncthreads();

    const int half = lane >> 4;   // 0: lanes 0-15, 1: lanes 16-31
    const int lid  = lane & 15;
    const int n0   = wave * 32;   // this wave's first output column

    v8f acc0 = {};
    v8f acc1 = {};

    #pragma unroll 4
    for (int k0 = 0; k0 < FEAT; k0 += 4) {
        const int ka = k0 + half * 2;           // K pair owned by this lane half
        // A fragment (16x4): lane half 0 -> K={k0,k0+1}, half 1 -> K={k0+2,k0+3}
        v2f a = *(const v2f*)&atile[lid][ka];
        // B fragments (4x16): N = n0+lid (and +16), same K pair as A
        const float* wp = W + (long)ka * FEAT + n0 + lid;
        v2f b0, b1;
        b0.x = wp[0];        b0.y = wp[FEAT];
        b1.x = wp[16];       b1.y = wp[FEAT + 16];
        acc0 = __builtin_amdgcn_wmma_f32_16x16x4_f32(false, a, false, b0,
                                                     (short)0, acc0, false, false);
        acc1 = __builtin_amdgcn_wmma_f32_16x16x4_f32(false, a, false, b1,
                                                     (short)0, acc1, false, false);
    }

    // D layout: VGPR i -> row i (lanes 0-15) / row i+8 (lanes 16-31), col = lane&15
    const float bias0 = bias[n0 + lid];
    const float bias1 = bias[n0 + 16 + lid];
    #pragma unroll
    for (int i = 0; i < 8; ++i) {
        const int r = i + half * 8;
        float* dst = out + (rowbase + r) * FEAT;
        dst[n0 + lid]      = acc0[i] + bias0;
        dst[n0 + 16 + lid] = acc1[i] + bias1;
    }
}

// ---------------------------------------------------------------------------
// Aggregation-buffer init with fused self-loop: out[n,:] = dinv[n]^2 * proj[n,:]
// ---------------------------------------------------------------------------
__global__ __launch_bounds__(256)
void selfloop_init(const float* __restrict__ proj, const float* __restrict__ dinv,
                   float* __restrict__ out) {
    long gid = (long)blockIdx.x * blockDim.x + threadIdx.x;   // N * 32 threads
    int n = (int)(gid >> 5);
    int l = (int)(gid & 31);
    if (n >= N_NODES) return;
    float d = dinv[n];
    float w = d * d;
    const float4 v = *(const float4*)(proj + (long)n * FEAT + l * 4);
    float4 r; r.x = w * v.x; r.y = w * v.y; r.z = w * v.z; r.w = w * v.w;
    *(float4*)(out + (long)n * FEAT + l * 4) = r;
}

// ---------------------------------------------------------------------------
// Edge scatter: out[c,:] += dinv[r]*dinv[c] * proj[r,:]  (1 wave per edge,
// float4 gather per lane, f32 global atomics do the RMW in L2)
// ---------------------------------------------------------------------------
__global__ __launch_bounds__(256)
void edge_scatter(const int* __restrict__ rows, const int* __restrict__ cols,
                  const float* __restrict__ dinv, const float* __restrict__ proj,
                  float* __restrict__ out, int e) {
    long gid = (long)blockIdx.x * blockDim.x + threadIdx.x;   // E * 32 threads
    int ei = (int)(gid >> 5);
    int l  = (int)(gid & 31);
    if (ei >= e) return;
    int r = rows[ei];
    int c = cols[ei];
    float w = dinv[r] * dinv[c];
    const float4 v = *(const float4*)(proj + (long)r * FEAT + l * 4);
    float* dst = out + (long)c * FEAT + l * 4;
    atomicAdd(dst + 0, w * v.x);
    atomicAdd(dst + 1, w * v.y);
    atomicAdd(dst + 2, w * v.z);
    atomicAdd(dst + 3, w * v.w);
}

__global__ __launch_bounds__(256)
void relu_inplace(float* __restrict__ h) {
    long gid = (long)blockIdx.x * blockDim.x + threadIdx.x;   // N * 32 threads
    if (gid >= (long)N_NODES * 32) return;
    float4 v = *(const float4*)(h + gid * 4);
    v.x = fmaxf(v.x, 0.f); v.y = fmaxf(v.y, 0.f);
    v.z = fmaxf(v.z, 0.f); v.w = fmaxf(v.w, 0.f);
    *(float4*)(h + gid * 4) = v;
}

// ---------------------------------------------------------------------------
// Global mean pool (scatter phase): gsum[batch[n],:] += h[n,:]; gcnt[g] += 1
// ---------------------------------------------------------------------------
__global__ __launch_bounds__(256)
void pool_scatter(const float* __restrict__ h, const int* __restrict__ batch,
                  float* __restrict__ gsum, float* __restrict__ gcnt) {
    long gid = (long)blockIdx.x * blockDim.x + threadIdx.x;   // N * 32 threads
    int n = (int)(gid >> 5);
    int l = (int)(gid & 31);
    if (n >= N_NODES) return;
    int g = batch[n];
    const float4 v = *(const float4*)(h + (long)n * FEAT + l * 4);
    float* dst = gsum + (long)g * FEAT + l * 4;
    atomicAdd(dst + 0, v.x);
    atomicAdd(dst + 1, v.y);
    atomicAdd(dst + 2, v.z);
    atomicAdd(dst + 3, v.w);
    if (l == 0) atomicAdd(&gcnt[g], 1.0f);
}

// ---------------------------------------------------------------------------
// Head: out[g] = relu(mean_g @ hw1 + hb1) @ hw2 + hb2   (one block per graph)
// ---------------------------------------------------------------------------
__global__ __launch_bounds__(128)
void head_mlp(const float* __restrict__ gsum, const float* __restrict__ gcnt,
              const float* __restrict__ hw1, const float* __restrict__ hb1,
              const float* __restrict__ hw2, const float* __restrict__ hb2,
              float* __restrict__ out) {
    __shared__ float gv[FEAT];
    __shared__ float red[FEAT];
    const int g = blockIdx.x;
    const int j = threadIdx.x;
    const float inv = 1.0f / fmaxf(gcnt[g], 1.0f);
    gv[j] = gsum[(long)g * FEAT + j] * inv;
    __syncthreads();
    float acc = hb1[j];
    #pragma unroll 8
    for (int k = 0; k < FEAT; ++k) acc += gv[k] * hw1[k * FEAT + j];
    red[j] = fmaxf(acc, 0.f) * hw2[j];
    __syncthreads();
    #pragma unroll
    for (int s = 64; s > 0; s >>= 1) {
        if (j < s) red[j] += red[j + s];
        __syncthreads();
    }
    if (j == 0) out[g] = red[0] + hb2[0];
}

// ---------------------------------------------------------------------------
// Launch
// ---------------------------------------------------------------------------
extern "C" void kernel_launch(void* const* d_in, const int* in_sizes, int n_in,
                              void* d_out, int out_size, void* d_ws, size_t ws_size,
                              hipStream_t stream) {
    const float* x    = (const float*)d_in[0];
    const int*   ei   = (const int*)d_in[1];     // [2, E] flat: row = ei, col = ei+E
    const int*   bat  = (const int*)d_in[2];
    const float* w1   = (const float*)d_in[3];
    const float* b1   = (const float*)d_in[4];
    const float* w2   = (const float*)d_in[5];
    const float* b2   = (const float*)d_in[6];
    const float* w3   = (const float*)d_in[7];
    const float* b3   = (const float*)d_in[8];
    const float* hw1  = (const float*)d_in[9];
    const float* hb1  = (const float*)d_in[10];
    const float* hw2  = (const float*)d_in[11];
    const float* hb2  = (const float*)d_in[12];
    float* out = (float*)d_out;

    const int* rows = ei;
    const int* cols = ei + N_EDGES;

    // workspace layout (256B aligned)
    char* ws = (char*)d_ws;
    size_t off = 0;
    auto take = [&](size_t bytes) { char* p = ws + off; off = (off + bytes + 255) & ~(size_t)255; return p; };
    float* dinv = (float*)take((size_t)N_NODES * sizeof(float));
    float* bufA = (float*)take((size_t)N_NODES * FEAT * sizeof(float));  // activations
    float* bufB = (float*)take((size_t)N_NODES * FEAT * sizeof(float));  // projection scratch
    float* gsum = (float*)take((size_t)NGRAPH * FEAT * sizeof(float));
    float* gcnt = (float*)take((size_t)NGRAPH * sizeof(float));
    (void)ws_size; (void)in_sizes; (void)n_in; (void)out_size;

    const int TB = 256;
    const int nwarp_blocks = (N_NODES * 32 + TB - 1) / TB;     // N*32-thread kernels
    const int ewarp_blocks = (int)(((long)N_EDGES * 32 + TB - 1) / TB);

    // --- degrees / norm (shared by all 3 layers) ---
    fill_f32<<<(N_NODES + TB - 1) / TB, TB, 0, stream>>>(dinv, 1.0f, N_NODES); // self loop
    deg_count<<<(N_EDGES + TB - 1) / TB, TB, 0, stream>>>(cols, dinv, N_EDGES);
    deg_to_dinv<<<(N_NODES + TB - 1) / TB, TB, 0, stream>>>(dinv, N_NODES);

    const float* layer_in = x;
    const float* Ws[3] = { w1, w2, w3 };
    const float* Bs[3] = { b1, b2, b3 };
    for (int l = 0; l < 3; ++l) {
        // proj = in @ W + b   (WMMA)
        gcn_gemm_wmma<<<N_NODES / 16, 128, 0, stream>>>(layer_in, Ws[l], Bs[l], bufB);
        // agg = dinv^2 * proj  (self loops) ...
        selfloop_init<<<nwarp_blocks, TB, 0, stream>>>(bufB, dinv, bufA);
        // ... += scatter over edges
        edge_scatter<<<ewarp_blocks, TB, 0, stream>>>(rows, cols, dinv, bufB, bufA, N_EDGES);
        relu_inplace<<<nwarp_blocks, TB, 0, stream>>>(bufA);
        layer_in = bufA;
    }

    // --- mean pool + head ---
    fill_f32<<<(NGRAPH * FEAT + TB - 1) / TB, TB, 0, stream>>>(gsum, 0.0f, NGRAPH * FEAT);
    fill_f32<<<1, NGRAPH, 0, stream>>>(gcnt, 0.0f, NGRAPH);
    pool_scatter<<<nwarp_blocks, TB, 0, stream>>>(bufA, bat, gsum, gcnt);
    head_mlp<<<NGRAPH, FEAT, 0, stream>>>(gsum, gcnt, hw1, hb1, hw2, hb2, out);
}